// FrequencyShiftMethod_24120536334971
// MI455X (gfx1250) — compile-verified
//
#include <hip/hip_runtime.h>

#ifndef __has_builtin
#define __has_builtin(x) 0
#endif

#define F_DIM   257
#define T_DIM   4096
#define ROWS    16            // staged input rows per tile
#define OUT_ROWS 15           // output rows per tile (row f needs rows f, f+1)
#define TCOLS   64            // float4 columns per tile (256 floats = 1 KB/row)
#define THREADS 256

typedef float fp32x4 __attribute__((ext_vector_type(4)));
typedef int   i32x4  __attribute__((ext_vector_type(4)));

#if defined(__gfx1250__) && __has_builtin(__builtin_amdgcn_global_load_async_to_lds_b128)
#define HAVE_ASYNC_LDS 1
#else
#define HAVE_ASYNC_LDS 0
#endif

#if HAVE_ASYNC_LDS
// 16-byte async global->LDS copy: lowers to global_load_async_to_lds_b128 (ASYNCcnt)
__device__ __forceinline__ void async_copy16(const float* g, fp32x4* l) {
    float* gnc = const_cast<float*>(g);                                   // drop const first
    __attribute__((address_space(1))) i32x4* gp =
        (__attribute__((address_space(1))) i32x4*)gnc;
    __attribute__((address_space(3))) i32x4* lp =
        (__attribute__((address_space(3))) i32x4*)l;
    __builtin_amdgcn_global_load_async_to_lds_b128(gp, lp, /*imm_offset=*/0, /*cpol=*/0);
}
#endif

// out = log10(beta*10^lo + alpha*10^hi + eps), via native exp2/log2 (v_exp_f32 / v_log_f32)
__device__ __forceinline__ float shift_one(float lo, float hi, float beta, float alpha) {
    const float LOG2_10  = 3.3219280948873623f;   // log2(10)
    const float RLOG2_10 = 0.3010299956639812f;   // 1/log2(10)
    float ea = __builtin_amdgcn_exp2f(lo * LOG2_10);
    float eb = __builtin_amdgcn_exp2f(hi * LOG2_10);
    float s  = fmaf(beta, ea, alpha * eb) + 1e-8f;
    return __builtin_amdgcn_logf(s) * RLOG2_10;
}

__global__ __launch_bounds__(THREADS)
void FrequencyShift_kernel(const float* __restrict__ x, float* __restrict__ out) {
    __shared__ fp32x4 smem[ROWS][TCOLS];          // 16 KB

    const int tid = threadIdx.x;
    const int t0  = blockIdx.x * (TCOLS * 4);     // time offset (floats)
    const int f0  = blockIdx.y * OUT_ROWS;        // first output frequency row
    const int b   = blockIdx.z;
    const size_t batch_base = (size_t)b * F_DIM * T_DIM;

    // ---- Stage ROWS input rows [f0 .. f0+15] (freq-clipped) into LDS ----
    // 1024 float4 transfers == 4 per thread, each a 16-byte async copy.
#pragma unroll
    for (int j = 0; j < (ROWS * TCOLS) / THREADS; ++j) {
        int i     = tid + j * THREADS;
        int row   = i >> 6;
        int col   = i & 63;
        int rclip = f0 + row;
        if (rclip > F_DIM - 1) rclip = F_DIM - 1;
        const float* g = x + batch_base + (size_t)rclip * T_DIM + t0 + col * 4;
#if HAVE_ASYNC_LDS
        async_copy16(g, &smem[row][col]);
#else
        smem[row][col] = *(const fp32x4*)g;
#endif
    }

#if HAVE_ASYNC_LDS
#if __has_builtin(__builtin_amdgcn_s_wait_asynccnt)
    __builtin_amdgcn_s_wait_asynccnt(0);
#else
    asm volatile("s_wait_asynccnt 0x0" ::: "memory");
#endif
#endif
    __syncthreads();

    // ---- Compute 15 output rows from LDS ----
    const int col = tid & 63;      // float4 column
    const int rg  = tid >> 6;      // row group 0..3; rows rg, rg+4, rg+8, rg+12
#pragma unroll
    for (int k = 0; k < 4; ++k) {
        int row = rg + 4 * k;
        if (row >= OUT_ROWS) break;            // only rg==3,k==3 trips this
        int f = f0 + row;
        if (f >= F_DIM) continue;              // tail F-tile

        float* o = out + batch_base + (size_t)f * T_DIM + t0 + col * 4;
        fp32x4 r;
        if (f == F_DIM - 1) {
            // ceil-source bin is out of range -> shifted = 0 -> log10(eps) = -8
            r.x = r.y = r.z = r.w = -8.0f;
        } else {
            // per-row fp32 alpha, matching the reference's rounding exactly
            float tgt   = (float)f + 0.64f;
            float alpha = tgt - floorf(tgt);
            float beta  = 1.0f - alpha;
            fp32x4 a = smem[row][col];
            fp32x4 h = smem[row + 1][col];
            r.x = shift_one(a.x, h.x, beta, alpha);
            r.y = shift_one(a.y, h.y, beta, alpha);
            r.z = shift_one(a.z, h.z, beta, alpha);
            r.w = shift_one(a.w, h.w, beta, alpha);
        }
        *(fp32x4*)o = r;                       // global_store_b128
    }
}

extern "C" void kernel_launch(void* const* d_in, const int* in_sizes, int n_in,
                              void* d_out, int out_size, void* d_ws, size_t ws_size,
                              hipStream_t stream) {
    const float* x  = (const float*)d_in[0];
    float*      out = (float*)d_out;

    const int B = in_sizes[0] / (F_DIM * T_DIM);   // 32 for the reference shapes
    dim3 grid(T_DIM / (TCOLS * 4),                 // 16 time tiles
              (F_DIM + OUT_ROWS - 1) / OUT_ROWS,   // 18 frequency tiles
              B);
    FrequencyShift_kernel<<<grid, THREADS, 0, stream>>>(x, out);
}